// MyLayer_64656437674424
// MI455X (gfx1250) — compile-verified
//
#include <hip/hip_runtime.h>
#include <cstdint>
#include <cstddef>

typedef __attribute__((ext_vector_type(2))) float v2f;
typedef __attribute__((ext_vector_type(8))) float v8f;

#define TM 128
#define TK 32
#define TN 128

// Async DMA of 16 bytes from global memory into LDS (CDNA5 gfx1250 path).
// Generic shared-pointer low 32 bits are the LDS byte offset (flat aperture rule).
__device__ __forceinline__ void async_b128(const float* g, const float* l) {
    unsigned      lds = (unsigned)(unsigned long long)l;
    unsigned long long ga = (unsigned long long)g;
    asm volatile("global_load_async_to_lds_b128 %0, %1, off"
                 :: "v"(lds), "v"(ga) : "memory");
}

// Native fp32 atomic add (no return) at device scope: RMW executes in L2's
// atomic units, the coherence point shared by all WGPs. Tracked by STOREcnt;
// S_ENDPGM's implicit wait-idle drains it before the next kernel.
__device__ __forceinline__ void gatomic_fadd(float* p, float v) {
    asm volatile("global_atomic_add_f32 %0, %1, off scope:SCOPE_DEV"
                 :: "v"((unsigned long long)p), "v"(v) : "memory");
}

// C[row, colOff + n] = rowscale[row] * A[row, :] @ W_rel[:, n]  (+ bias[n])
// W_rel = Wbase + blockIdx.y * K * 128   (row-major [K,128])
// One block = 128 rows x 128 cols; 8 waves, each wave 16 rows x 128 cols.
__global__ __launch_bounds__(256) void gemm_rowscale_wmma(
    const float* __restrict__ A, int lda,
    const float* __restrict__ rowscale,
    const float* __restrict__ Wbase, int K,
    const float* __restrict__ bias,
    float* __restrict__ Cout, int ldc,
    int nrows)
{
    __shared__ float sA[TM * TK];   // [m][k]
    __shared__ float sB[TK * TN];   // [k][n]

    const int tid    = threadIdx.x;
    const int wave   = tid >> 5;
    const int lane   = tid & 31;
    const int lhalf  = lane & 15;
    const bool hiH   = lane >= 16;
    const int rel    = blockIdx.y;
    const float* W   = Wbase + (size_t)rel * K * TN;
    const int colOff = rel * TN;
    const int m0     = blockIdx.x * TM;
    const int mW     = wave * 16;

    v8f acc[8] = {};

    for (int k0 = 0; k0 < K; k0 += TK) {
        // ---- A tile: predicated loads, row-scaled on the way in ----
        #pragma unroll
        for (int i = 0; i < 16; ++i) {          // 128*32 / 256 threads
            int e = i * 256 + tid;
            int m = e >> 5;                     // / TK
            int k = e & (TK - 1);
            int row = m0 + m;
            float v = 0.0f;
            if (row < nrows) v = A[(size_t)row * lda + k0 + k] * rowscale[row];
            sA[m * TK + k] = v;
        }
        // ---- B tile: async DMA to LDS (always in-bounds) ----
        #pragma unroll
        for (int i = 0; i < 4; ++i) {           // 32*128 floats / (256 thr * 4)
            int f  = i * 256 + tid;             // float4 id
            int k  = f >> 5;                    // 32 float4 per k-row
            int n4 = f & 31;
            async_b128(W + (size_t)(k0 + k) * TN + n4 * 4,
                       &sB[k * TN + n4 * 4]);
        }
        asm volatile("s_wait_asynccnt 0" ::: "memory");
        __syncthreads();

        // ---- 8 K-steps of 4, 8 N-subtiles each: 64 WMMAs per chunk ----
        #pragma unroll
        for (int kk = 0; kk < TK; kk += 4) {
            const int kb = kk + (hiH ? 2 : 0);
            v2f a;
            a.x = sA[(mW + lhalf) * TK + kb];
            a.y = sA[(mW + lhalf) * TK + kb + 1];
            #pragma unroll
            for (int nt = 0; nt < 8; ++nt) {
                v2f b;
                b.x = sB[kb * TN + nt * 16 + lhalf];
                b.y = sB[(kb + 1) * TN + nt * 16 + lhalf];
                acc[nt] = __builtin_amdgcn_wmma_f32_16x16x4_f32(
                    false, a, false, b, (short)0, acc[nt], false, false);
            }
        }
        __syncthreads();
    }

    // ---- epilogue: C/D layout -> global ----
    #pragma unroll
    for (int nt = 0; nt < 8; ++nt) {
        int col  = nt * 16 + lhalf;
        float bv = bias ? bias[col] : 0.0f;
        #pragma unroll
        for (int j = 0; j < 8; ++j) {
            int row = m0 + mW + j + (hiH ? 8 : 0);
            if (row < nrows)
                Cout[(size_t)row * ldc + colOff + col] = acc[nt][j] + bv;
        }
    }
}

// One wave32 per (relation, edge): gather 128 floats (4/lane) and
// atomic-add them to the destination node row; both directions.
__global__ __launch_bounds__(256) void scatter_edges(
    const float* __restrict__ xu, const float* __restrict__ xi,
    const int* __restrict__ eu, const int* __restrict__ ei,
    float* __restrict__ hu, float* __restrict__ hi,
    int E, int R)
{
    long long gtid = (long long)blockIdx.x * blockDim.x + threadIdx.x;
    long long w    = gtid >> 5;
    int lane       = threadIdx.x & 31;
    long long tot  = (long long)E * R;
    if (w >= tot) return;
    int r = (int)(w / E);
    int e = (int)(w - (long long)r * E);
    int u  = eu[(size_t)r * E + e];
    int it = ei[(size_t)r * E + e];
    size_t su = (size_t)u  * 640 + r * 128 + lane * 4;
    size_t si = (size_t)it * 640 + r * 128 + lane * 4;

    float4 a = *(const float4*)(xu + su);       // user -> item
    float* di = hi + si;
    gatomic_fadd(di + 0, a.x); gatomic_fadd(di + 1, a.y);
    gatomic_fadd(di + 2, a.z); gatomic_fadd(di + 3, a.w);

    float4 b = *(const float4*)(xi + si);       // item -> user
    float* du = hu + su;
    gatomic_fadd(du + 0, b.x); gatomic_fadd(du + 1, b.y);
    gatomic_fadd(du + 2, b.z); gatomic_fadd(du + 3, b.w);
}

extern "C" void kernel_launch(void* const* d_in, const int* in_sizes, int n_in,
                              void* d_out, int out_size, void* d_ws, size_t ws_size,
                              hipStream_t stream) {
    (void)in_sizes; (void)n_in; (void)out_size; (void)ws_size;
    const float* ufeat   = (const float*)d_in[0];
    const float* ifeat   = (const float*)d_in[1];
    const float* cj_user = (const float*)d_in[2];
    const float* ci_user = (const float*)d_in[3];
    const float* cj_item = (const float*)d_in[4];
    const float* ci_item = (const float*)d_in[5];
    const float* W_u     = (const float*)d_in[6];
    const float* W_i     = (const float*)d_in[7];
    const float* Wu_fc   = (const float*)d_in[8];
    const float* bu      = (const float*)d_in[9];
    const float* Wi_fc   = (const float*)d_in[10];
    const float* bi      = (const float*)d_in[11];
    const int*   edge_u  = (const int*)d_in[12];
    const int*   edge_i  = (const int*)d_in[13];

    const int NU = 100000, NI = 50000, RDIM = 5, EDG = 1000000;
    const int RM = 640;                 // R * MSG

    float* xu = (float*)d_ws;           // [NU, 640]
    float* xi = xu + (size_t)NU * RM;   // [NI, 640]
    float* hu = xi + (size_t)NI * RM;   // [NU, 640] accumulators
    float* hi = hu + (size_t)NU * RM;   // [NI, 640]

    // zero scatter accumulators every call (graph-capture safe)
    hipMemsetAsync(hu, 0, ((size_t)NU * RM + (size_t)NI * RM) * sizeof(float), stream);

    dim3 blk(256);
    // message GEMMs: x = (cj . feat) @ W[r]
    gemm_rowscale_wmma<<<dim3((NU + TM - 1) / TM, RDIM), blk, 0, stream>>>(
        ufeat, 256, cj_user, W_u, 256, nullptr, xu, RM, NU);
    gemm_rowscale_wmma<<<dim3((NI + TM - 1) / TM, RDIM), blk, 0, stream>>>(
        ifeat, 256, cj_item, W_i, 256, nullptr, xi, RM, NI);

    // edge scatter (both directions per edge)
    long long waves  = (long long)RDIM * EDG;
    int       blocks = (int)((waves * 32 + 255) / 256);
    scatter_edges<<<blocks, blk, 0, stream>>>(xu, xi, edge_u, edge_i, hu, hi, EDG, RDIM);

    // FC GEMMs: out = (ci . h) @ W_fc + b
    float* out_u = (float*)d_out;
    float* out_i = out_u + (size_t)NU * 128;
    gemm_rowscale_wmma<<<dim3((NU + TM - 1) / TM, 1), blk, 0, stream>>>(
        hu, RM, ci_user, Wu_fc, RM, bu, out_u, 128, NU);
    gemm_rowscale_wmma<<<dim3((NI + TM - 1) / TM, 1), blk, 0, stream>>>(
        hi, RM, ci_item, Wi_fc, RM, bi, out_i, 128, NI);
}